// ReactionEncoder_75711683494310
// MI455X (gfx1250) — compile-verified
//
#include <hip/hip_runtime.h>

// ---------------------------------------------------------------------------
// ReactionEncoder — pure data-movement op (no matmul):
//   out = concat(diff_atom[81920,256], diff_bond[98304,256], diff_global[2048,256])
// Roofline: ~540 MB DRAM traffic vs ~0.1 GFLOP (0.2 FLOP/byte) -> limited by
// 23.3 TB/s HBM (~23 us). WMMA does not apply (segment-sum K<=2; bond matrix
// <=2 nnz/row -> dense formulation moves the same bytes). The CDNA5 feature
// that matches this op is the async global->LDS DMA path (ASYNCcnt): one of
// diff_atom's two input streams rides the async DMA engine while the other
// rides the normal vmem-return path, doubling per-wave memory-level
// parallelism across independent hardware queues.
// ---------------------------------------------------------------------------

typedef float v4f __attribute__((ext_vector_type(4)));
typedef int   v4i __attribute__((vector_size(16)));  // matches builtin proto

static constexpr int D  = 256;  // feature dim (f32)
static constexpr int D4 = 64;   // float4s per row

#if __has_builtin(__builtin_amdgcn_global_load_async_to_lds_b128) && \
    __has_builtin(__builtin_amdgcn_s_wait_asynccnt)
#define USE_ASYNC_LDS 1
#else
#define USE_ASYNC_LDS 0
#endif

#define AS1 __attribute__((address_space(1)))
#define AS3 __attribute__((address_space(3)))

// ---------------------------------------------------------------------------
// Kernel 1: diff_atom = atom[half + i] - atom[i], streaming subtract.
// Branch-free: both the load index and the store index are clamped to n4-1,
// so tail lanes (never present here: n4 % 256 == 0) would just recompute and
// re-store element n4-1's identical value -- in-bounds, race-free-by-value,
// deterministic. No control flow means the compiler cannot sink the hi-half
// vmem load behind s_wait_asynccnt: issue order is
//   global_load_async_to_lds_b128 (ASYNCcnt)  -> lo half into LDS
//   global_load_b128 th:NT        (LOADcnt)   -> hi half into VGPRs
//   s_wait_asynccnt 0 ; ds_load_b128 ; v_pk_add_f32 ; global_store_b128 th:NT
// Each lane reads back only its own LDS slot, so the wave-level asynccnt wait
// is sufficient ordering -- no workgroup barrier needed.
// ---------------------------------------------------------------------------
__global__ __launch_bounds__(256) void diff_atom_kernel(
    const float* __restrict__ atom, float* __restrict__ out, long n4) {
#if USE_ASYNC_LDS
  __shared__ v4f stage[256];
  const int  tid = threadIdx.x;
  const long gid = (long)blockIdx.x * 256 + tid;
  const long idx = (gid < n4) ? gid : (n4 - 1);  // clamp loads AND store

  const v4f* lo = (const v4f*)atom;
  const v4f* hi = lo + n4;

  // CDNA5 async DMA: copies 16 bytes from global lo[idx] into LDS slot stage[tid]
  __builtin_amdgcn_global_load_async_to_lds_b128(
      (AS1 v4i*)(lo + idx),
      (AS3 v4i*)(&stage[tid]),
      /*offset=*/0, /*cpol=*/0);

  // Overlap: hi-half rides the normal vmem path meanwhile.
  v4f a = __builtin_nontemporal_load(&hi[idx]);

  __builtin_amdgcn_s_wait_asynccnt(0);  // s_wait_asynccnt 0
  v4f b = stage[tid];

  __builtin_nontemporal_store(a - b, (v4f*)out + idx);
#else
  long gid = (long)blockIdx.x * blockDim.x + threadIdx.x;
  if (gid >= n4) return;
  const v4f* lo = (const v4f*)atom;
  const v4f* hi = lo + n4;
  v4f a = __builtin_nontemporal_load(&hi[gid]);
  v4f b = __builtin_nontemporal_load(&lo[gid]);
  v4f d = a - b;
  __builtin_nontemporal_store(d, (v4f*)out + gid);
#endif
}

// ---------------------------------------------------------------------------
// Kernel 2: diff_bond[row] = coef_p[row]*p_b[idx_p[row]] + coef_r[row]*r_b[idx_r[row]]
// coefs are in {-1,0,+1}; skip the load entirely when coef==0 (wave-uniform
// branch: 64 threads/row, wave32 -> each wave sees exactly one row). With the
// skip, exactly 88 of 96 candidate rows are read per reaction -- the same
// bytes a dense-matmul formulation would move, at none of its layout cost.
// ---------------------------------------------------------------------------
__global__ __launch_bounds__(256) void diff_bond_kernel(
    const float* __restrict__ bond,
    const int* __restrict__ idx_r, const int* __restrict__ idx_p,
    const int* __restrict__ coef_r, const int* __restrict__ coef_p,
    float* __restrict__ out, long n_rows, long react_rows) {
  long gid = (long)blockIdx.x * blockDim.x + threadIdx.x;
  if (gid >= n_rows * D4) return;
  long row = gid >> 6;          // output bond row
  int  c   = (int)(gid & 63);   // float4 column within row

  const v4f* rb = (const v4f*)bond;            // reactant bond rows
  const v4f* pb = rb + react_rows * (long)D4;  // product bond rows

  int cp = coef_p[row];
  int cr = coef_r[row];

  v4f acc = {0.f, 0.f, 0.f, 0.f};
  if (cp != 0) {
    long ip = (long)idx_p[row];
    v4f v = __builtin_nontemporal_load(&pb[ip * D4 + c]);
    acc += (float)cp * v;
  }
  if (cr != 0) {
    long ir = (long)idx_r[row];
    v4f v = __builtin_nontemporal_load(&rb[ir * D4 + c]);
    acc += (float)cr * v;
  }
  __builtin_nontemporal_store(acc, (v4f*)out + gid);
}

// ---------------------------------------------------------------------------
// Kernel 3: diff_global[b] = sum_j gp[b*PM+j] - sum_j gr[b*RM+j]
// seg_r/seg_p are repeat(arange(B), RM/PM) -> segments are contiguous,
// so the segment-sum is a direct strided reduction (no atomics needed).
// ---------------------------------------------------------------------------
__global__ __launch_bounds__(256) void diff_global_kernel(
    const float* __restrict__ gf, float* __restrict__ out,
    long num_react, int RM, int PM, long react_mols) {
  long gid = (long)blockIdx.x * blockDim.x + threadIdx.x;
  if (gid >= num_react * D4) return;
  long b = gid >> 6;
  int  c = (int)(gid & 63);

  const v4f* gr = (const v4f*)gf;
  const v4f* gp = gr + react_mols * (long)D4;

  v4f acc = {0.f, 0.f, 0.f, 0.f};
  for (int j = 0; j < PM; ++j)
    acc += __builtin_nontemporal_load(&gp[(b * PM + j) * (long)D4 + c]);
  for (int j = 0; j < RM; ++j)
    acc -= __builtin_nontemporal_load(&gr[(b * RM + j) * (long)D4 + c]);

  __builtin_nontemporal_store(acc, (v4f*)out + gid);
}

// ---------------------------------------------------------------------------
// Host launcher. All shape parameters derived from in_sizes/out_size so we
// never have to read device-side scalars (graph-capture safe, deterministic).
//   in[0] atom_feats  (2*B*ATOMS, D)      in[5] coef_r (B*OUT_B)
//   in[1] bond_feats  (B*(RB+PB), D)      in[6] coef_p (B*OUT_B)
//   in[2] global_feats(B*(RM+PM), D)      in[7] seg_r  (B*RM)
//   in[3] idx_r       (B*OUT_B)           in[8] seg_p  (B*PM)
//   in[4] idx_p       (B*OUT_B)           in[9..12] scalars (unused on host)
// ---------------------------------------------------------------------------
extern "C" void kernel_launch(void* const* d_in, const int* in_sizes, int n_in,
                              void* d_out, int out_size, void* d_ws, size_t ws_size,
                              hipStream_t stream) {
  const float* atom  = (const float*)d_in[0];
  const float* bond  = (const float*)d_in[1];
  const float* glob  = (const float*)d_in[2];
  const int* idx_r   = (const int*)d_in[3];
  const int* idx_p   = (const int*)d_in[4];
  const int* coef_r  = (const int*)d_in[5];
  const int* coef_p  = (const int*)d_in[6];

  const long half_rows       = (long)in_sizes[0] / (2 * D);  // B*ATOMS   = 81920
  const long out_bond_rows   = (long)in_sizes[3];            // B*OUT_B   = 98304
  const long total_bond_rows = (long)in_sizes[1] / D;        // B*(RB+PB) = 180224
  const long react_bond_rows = total_bond_rows / 2;          // RB == PB  -> 90112
  const long react_mols      = (long)in_sizes[7];            // B*RM      = 4096
  // num_reactions follows from the output layout: out rows = atoms + bonds + B
  const long num_react = (long)out_size / D - half_rows - out_bond_rows;  // 2048
  const int  RM = (int)(react_mols / num_react);             // 2
  const int  PM = (int)((long)in_sizes[8] / num_react);      // 1

  float* out_atom = (float*)d_out;
  float* out_bond = out_atom + half_rows * (long)D;
  float* out_glob = out_bond + out_bond_rows * (long)D;

  const int block = 256;

  {  // diff_atom: 5.24M float4s (n4 % 256 == 0 here; clamp handles any tail)
    long n4 = half_rows * (long)D4;
    long grid = (n4 + block - 1) / block;
    diff_atom_kernel<<<dim3((unsigned)grid), dim3(block), 0, stream>>>(
        atom, out_atom, n4);
  }
  {  // diff_bond: 6.29M float4s
    long n = out_bond_rows * (long)D4;
    long grid = (n + block - 1) / block;
    diff_bond_kernel<<<dim3((unsigned)grid), dim3(block), 0, stream>>>(
        bond, idx_r, idx_p, coef_r, coef_p, out_bond,
        out_bond_rows, react_bond_rows);
  }
  {  // diff_global: 131K float4s
    long n = num_react * (long)D4;
    long grid = (n + block - 1) / block;
    diff_global_kernel<<<dim3((unsigned)grid), dim3(block), 0, stream>>>(
        glob, out_glob, num_react, RM, PM, react_mols);
  }
}